// SoftCountFromSlices_56075093016618
// MI455X (gfx1250) — compile-verified
//
#include <hip/hip_runtime.h>
#include <hip/hip_bf16.h>
#include <stdint.h>

// ---------------------------------------------------------------------------
// SoftCountFromSlices (B=16384 rows, T=2048 steps, 17 bins).
// Pure bandwidth problem: 128 MiB @ 23.3 TB/s ~ 5.8 us floor, ~12.75 FLOP/B.
// Design: one wave32 workgroup per 32 rows (512 blocks -> fills WGPs, zero
// barriers), TDM async 2D tile loads (global->LDS) with a 6-deep ring of
// buffers (5 outstanding descriptors/wave => ~22 MB in flight device-wide,
// matching the HBM bandwidth-delay product). TDM pad feature gives an odd
// LDS row stride (65 dwords) -> bank-conflict-free ds_load_b32 consumption.
// ---------------------------------------------------------------------------

#define T_LEN     2048
#define CHUNK     64               // timesteps per staged tile
#define ROWS      32               // rows per block == blockDim.x == one wave32
#define NCHUNK    (T_LEN / CHUNK)  // 32
#define NBUF      6                // ring depth (5 prefetches outstanding)
#define ROWSTRIDE (CHUNK + 1)      // 65 floats after TDM padding (odd stride)
#define BUF_FLOATS (ROWS * ROWSTRIDE)  // 2080
#define NBINS     17

typedef unsigned int u32;
typedef __attribute__((ext_vector_type(4))) u32 v4u;
typedef __attribute__((ext_vector_type(8))) int v8i;
typedef __attribute__((ext_vector_type(4))) int v4i;

// Issue one TDM 2D tile load: [ROWS x CHUNK] f32, global row stride T_LEN,
// padded by 1 dword per CHUNK dwords in LDS (row stride ROWSTRIDE).
__device__ __forceinline__ void tdm_load_tile(const float* gptr, u32 lds_byte_off) {
    uint64_t ga = (uint64_t)(uintptr_t)gptr;

    // ---- D# group 0: control / lds_addr / global_addr / type ----
    v4u g0;
    g0.x = 1u;                                        // count=1 (valid), user mode
    g0.y = lds_byte_off;                              // lds_addr
    g0.z = (u32)ga;                                   // global_addr[31:0]
    g0.w = (u32)((ga >> 32) & 0x01FFFFFFu)            // global_addr[56:32]
         | (2u << 30);                                // type=2 ("image")

    // ---- D# group 1: sizes / strides / padding ----
    // [17:16] data_size=2 (4B), [20] pad_enable=1,
    // [24:22] pad_interval=5 (pad after every 256B = 64 dwords),
    // [31:25] pad_amount=0 (1 dword of padding)
    v8i g1;
    g1[0] = (int)((2u << 16) | (1u << 20) | (5u << 22));
    g1[1] = (int)(((u32)T_LEN & 0xFFFFu) << 16);      // tensor_dim0[15:0] = 2048
    g1[2] = (int)((((u32)T_LEN >> 16) & 0xFFFFu)      // tensor_dim0[31:16]
         | (((u32)ROWS & 0xFFFFu) << 16));            // tensor_dim1[15:0] = 32
    g1[3] = (int)(((u32)CHUNK) << 16);                // tensor_dim1[31:16]=0, tile_dim0=64
    g1[4] = (int)((u32)ROWS);                         // tile_dim1=32, tile_dim2=0
    g1[5] = (int)((u32)T_LEN);                        // tensor_dim0_stride = 2048
    g1[6] = 0;
    g1[7] = 0;

    v4i z4 = (v4i)0;                                  // groups 2/3 unused (2D tensor)
    v8i z8 = (v8i)0;

    __builtin_amdgcn_tensor_load_to_lds(g0, g1, z4, z4, z8, 0);
}

// Wait until at most `n` tensor ops outstanding; fence so subsequent LDS
// reads cannot be hoisted above the wait.
__device__ __forceinline__ void wait_tensor_le(int n) {
    switch (n) {
        case 0: __builtin_amdgcn_s_wait_tensorcnt(0); break;
        case 1: __builtin_amdgcn_s_wait_tensorcnt(1); break;
        case 2: __builtin_amdgcn_s_wait_tensorcnt(2); break;
        case 3: __builtin_amdgcn_s_wait_tensorcnt(3); break;
        case 4: __builtin_amdgcn_s_wait_tensorcnt(4); break;
        default: __builtin_amdgcn_s_wait_tensorcnt(5); break;
    }
    asm volatile("" ::: "memory");
}

__global__ __launch_bounds__(ROWS)
void SoftCountFromSlices_kernel(const float* __restrict__ probs,
                                float* __restrict__ out, int B) {
    __shared__ float tile[NBUF * BUF_FLOATS];         // 49,920 bytes

    const int tid  = (int)threadIdx.x;                // lane id (wave32 block)
    const int row0 = (int)blockIdx.x * ROWS;
    const int row  = row0 + tid;

    // Low 32 bits of a generic LDS pointer == workgroup-relative LDS byte addr.
    const u32 lds_base = (u32)(uintptr_t)(void*)&tile[0];
    const float* gbase = probs + (size_t)row0 * T_LEN;

    float dp[NBINS];
    dp[0] = 1.0f;
#pragma unroll
    for (int k = 1; k < NBINS; ++k) dp[k] = 0.0f;

    // Prime the ring: NBUF-1 chunks in flight. Uniform (scalar) issue; the
    // whole wave executes it once (TDM ignores EXEC, operands are SGPRs).
#pragma unroll
    for (int i = 0; i < NBUF - 1; ++i)
        tdm_load_tile(gbase + (size_t)i * CHUNK,
                      lds_base + (u32)(i * BUF_FLOATS * 4));

    for (int c = 0; c < NCHUNK; ++c) {
        const int nxt = c + NBUF - 1;
        if (nxt < NCHUNK) {
            // Refill the buffer consumed last iteration. Ensure its ds_loads
            // fully retired before the DMA overwrites it.
            asm volatile("s_wait_dscnt 0" ::: "memory");
            tdm_load_tile(gbase + (size_t)nxt * CHUNK,
                          lds_base + (u32)((nxt % NBUF) * BUF_FLOATS * 4));
        }
        // In-order TDM completion: chunk c is done once at most
        // min(NBUF-1, chunks newer than c) remain outstanding.
        int newer = NCHUNK - 1 - c;
        if (newer > NBUF - 1) newer = NBUF - 1;
        wait_tensor_le(newer);

        const float* rp = &tile[(c % NBUF) * BUF_FLOATS + tid * ROWSTRIDE];
#pragma unroll 4
        for (int t = 0; t < CHUNK; ++t) {
            const float p    = rp[t];
            const float pm   = 1.0f - p;
            const float last = dp[16];
#pragma unroll
            for (int k = NBINS - 1; k >= 1; --k)
                dp[k] = dp[k] * pm + dp[k - 1] * p;
            dp[0] = dp[0] * pm;
            dp[16] += last;   // dp_new[:, -1] += dp[:, -1]
        }
    }

    if (row < B) {
        float* op = out + (size_t)row * NBINS;
#pragma unroll
        for (int k = 0; k < NBINS; ++k) op[k] = dp[k];
    }
}

extern "C" void kernel_launch(void* const* d_in, const int* in_sizes, int n_in,
                              void* d_out, int out_size, void* d_ws, size_t ws_size,
                              hipStream_t stream) {
    const float* probs = (const float*)d_in[0];
    float* out = (float*)d_out;
    const int B = in_sizes[0] / T_LEN;       // 16384
    const int grid = (B + ROWS - 1) / ROWS;  // 512
    SoftCountFromSlices_kernel<<<grid, ROWS, 0, stream>>>(probs, out, B);
}